// GCELoss_20959440404671
// MI455X (gfx1250) — compile-verified
//
#include <hip/hip_runtime.h>
#include <hip/hip_bf16.h>

// GCE loss: per-row top-k(=C/4) softmax-subset cross-entropy.
// Design: 1 workgroup per row; entire 125KB row DMA'd global->LDS once
// (CDNA5 async-tensor path, GLOBAL_LOAD_ASYNC_TO_LDS_B128); all subsequent
// passes (max, 4x radix-select histogram, exp-sum) run out of LDS.
// Single HBM pass => ~11.3us floor at 23.3 TB/s for the 262MB input.

#define CLASS_NUM 32000
#define BATCH     2048
#define TOPK      (CLASS_NUM / 4)
#define NT        512

// dynamic LDS layout (bytes)
#define ROW_OFF   0
#define HIST_OFF  (CLASS_NUM * 4)            // 128000
#define RED_OFF   (HIST_OFF + 256 * 4)       // 129024
#define SCAL_OFF  (RED_OFF + NT * 4)         // 131072
#define SMEM_SZ   (SCAL_OFF + 16)            // 131088

#if defined(__has_builtin)
#if __has_builtin(__builtin_amdgcn_global_load_async_to_lds_b128)
#define USE_ASYNC_BUILTIN 1
#endif
#if __has_builtin(__builtin_amdgcn_s_wait_asynccnt)
#define USE_WAIT_BUILTIN 1
#endif
#endif

// builtin prototype (from clang diagnostic):
//   void __builtin_amdgcn_global_load_async_to_lds_b128(v4i AS1*, v4i AS3*, imm, imm)
typedef int v4i __attribute__((vector_size(4 * sizeof(int))));
typedef __attribute__((address_space(1))) v4i* as1_v4i_p;
typedef __attribute__((address_space(3))) v4i* as3_v4i_p;

// order-preserving float->uint key (larger float => larger key)
__device__ __forceinline__ unsigned mono_key(float f) {
    unsigned u = __float_as_uint(f);
    return (u & 0x80000000u) ? ~u : (u | 0x80000000u);
}
__device__ __forceinline__ float inv_key(unsigned k) {
    unsigned u = (k & 0x80000000u) ? (k & 0x7FFFFFFFu) : ~k;
    return __uint_as_float(u);
}

// 16B async copy global -> LDS (CDNA5 GLOBAL_LOAD_ASYNC_TO_LDS_B128)
__device__ __forceinline__ void async_copy_b128(const float* g, float* l) {
#if defined(USE_ASYNC_BUILTIN)
    __builtin_amdgcn_global_load_async_to_lds_b128((as1_v4i_p)g, (as3_v4i_p)l,
                                                   /*offset=*/0, /*cpol=*/0);
#else
    // vdst = VGPR holding LDS byte address; vaddr = 64-bit global addr (GV mode)
    unsigned lds_addr = (unsigned)(unsigned long long)l; // low 32b of flat LDS addr
    unsigned long long ga = (unsigned long long)g;
    asm volatile("global_load_async_to_lds_b128 %0, %1, off"
                 :: "v"(lds_addr), "v"(ga) : "memory");
#endif
}

__device__ __forceinline__ void wait_async0() {
#if defined(USE_WAIT_BUILTIN)
    __builtin_amdgcn_s_wait_asynccnt(0);
#else
    asm volatile("s_wait_asynccnt 0" ::: "memory");
#endif
}

extern "C" __global__ void __launch_bounds__(NT)
gce_row_kernel(const float* __restrict__ logits,
               const long long* __restrict__ labels,
               float* __restrict__ row_loss) {
    extern __shared__ unsigned char smem[];
    float*    row  = (float*)(smem + ROW_OFF);
    unsigned* hist = (unsigned*)(smem + HIST_OFF);
    float*    red  = (float*)(smem + RED_OFF);
    unsigned* sc   = (unsigned*)(smem + SCAL_OFF); // [0]=prefix, [1]=remk

    const int tid = threadIdx.x;
    const int r   = blockIdx.x;
    const float* g = logits + (size_t)r * CLASS_NUM;

    // ---- 1) async DMA entire row into LDS (8000 x B128) ----
    for (int s = tid; s < CLASS_NUM / 4; s += NT)
        async_copy_b128(g + 4 * s, row + 4 * s);
    wait_async0();
    __syncthreads();

    // ---- 2) row max (needed for stable exp) ----
    float lm = -__builtin_inff();
    for (int i = tid; i < CLASS_NUM; i += NT) lm = fmaxf(lm, row[i]);
    red[tid] = lm;
    __syncthreads();
    for (int s = NT / 2; s > 0; s >>= 1) {
        if (tid < s) red[tid] = fmaxf(red[tid], red[tid + s]);
        __syncthreads();
    }
    const float rmax = red[0];
    __syncthreads();

    // ---- 3) radix select: key of the k-th largest element ----
    unsigned prefix = 0, pmask = 0, remk = TOPK;
    for (int pass = 3; pass >= 0; --pass) {
        const int sh = pass * 8;
        for (int b = tid; b < 256; b += NT) hist[b] = 0;
        __syncthreads();
        for (int i = tid; i < CLASS_NUM; i += NT) {
            unsigned key = mono_key(row[i]);
            if ((key & pmask) == prefix)
                atomicAdd(&hist[(key >> sh) & 255u], 1u);
        }
        __syncthreads();
        if (tid == 0) {
            unsigned acc = 0, sel = 0, inc = remk;
            for (int b = 255; b >= 0; --b) {
                unsigned c = hist[b];
                if (acc + c >= remk) { sel = (unsigned)b; inc = remk - acc; break; }
                acc += c;
            }
            sc[0] = prefix | (sel << sh);
            sc[1] = inc;
        }
        __syncthreads();
        prefix = sc[0];
        remk   = sc[1];   // # threshold-valued elements included in top-k
        pmask |= 0xFFu << sh;
    }
    const unsigned thr = prefix;

    // ---- 4) sum exp over top-k (strictly-greater + included ties) ----
    float ls = 0.f;
    for (int i = tid; i < CLASS_NUM; i += NT) {
        float f = row[i];
        if (mono_key(f) > thr) ls += __expf(f - rmax);
    }
    red[tid] = ls;
    __syncthreads();
    for (int s = NT / 2; s > 0; s >>= 1) {
        if (tid < s) red[tid] += red[tid + s];
        __syncthreads();
    }

    if (tid == 0) {
        float tot = red[0] + (float)remk * __expf(inv_key(thr) - rmax);
        long long lab = labels[r];
        float fl = row[(int)lab];
        if (mono_key(fl) < thr) tot += __expf(fl - rmax); // label outside top-k
        row_loss[r] = -(fl - rmax) + __logf(tot);
    }
}

extern "C" __global__ void __launch_bounds__(256)
gce_reduce_kernel(const float* __restrict__ row_loss, float* __restrict__ out) {
    __shared__ float red[256];
    float s = 0.f;
    for (int i = threadIdx.x; i < BATCH; i += 256) s += row_loss[i];
    red[threadIdx.x] = s;
    __syncthreads();
    for (int t = 128; t > 0; t >>= 1) {
        if (threadIdx.x < t) red[threadIdx.x] += red[threadIdx.x + t];
        __syncthreads();
    }
    if (threadIdx.x == 0) out[0] = red[0] / (float)BATCH;
}

extern "C" void kernel_launch(void* const* d_in, const int* in_sizes, int n_in,
                              void* d_out, int out_size, void* d_ws, size_t ws_size,
                              hipStream_t stream) {
    const float*     logits = (const float*)d_in[0];
    const long long* labels = (const long long*)d_in[1]; // int64 labels
    float* row_loss = (float*)d_ws;                       // 2048 floats scratch

    gce_row_kernel<<<BATCH, NT, SMEM_SZ, stream>>>(logits, labels, row_loss);
    gce_reduce_kernel<<<1, 256, 0, stream>>>(row_loss, (float*)d_out);
}